// SAGE_48808008352171
// MI455X (gfx1250) — compile-verified
//
#include <hip/hip_runtime.h>

// ---------------------------------------------------------------------------
// GraphSAGE 2-layer forward for MI455X (gfx1250, wave32).
//
// Roofline: 3.3 GFLOP of GEMM vs ~hundreds of MB of edge traffic -> entirely
// bandwidth bound (23.3 TB/s HBM, 192 MB L2 holds the whole 25.6 MB feature
// table). Edge aggregation uses global f32 atomics (L2-resident); the node
// transform uses V_WMMA_F32_16X16X4_F32 so the matrix path stays in native
// f32 with degree-norm applied on the AGG operand load and bias + ReLU fused into the WMMA epilogue.
// ---------------------------------------------------------------------------

typedef __attribute__((ext_vector_type(2))) float v2f;
typedef __attribute__((ext_vector_type(8))) float v8f;

#define N_NODES 100000
#define N_EDGES 1200000
#define FEAT    64

// ---------------------------------------------------------------------------
// Zero a float region (grid-stride).
// ---------------------------------------------------------------------------
__global__ void sage_zero_kernel(float* __restrict__ p, long long n) {
    long long i = (long long)blockIdx.x * blockDim.x + threadIdx.x;
    long long stride = (long long)gridDim.x * blockDim.x;
    for (; i < n; i += stride) p[i] = 0.0f;
}

// ---------------------------------------------------------------------------
// deg[dst[e]] += 1  (computed once; identical for both layers).
// ---------------------------------------------------------------------------
__global__ void sage_degree_kernel(const int* __restrict__ dst,
                                   float* __restrict__ deg, int n_edges) {
    int e = blockIdx.x * blockDim.x + threadIdx.x;
    if (e < n_edges) atomicAdd(deg + dst[e], 1.0f);
}

// ---------------------------------------------------------------------------
// agg[dst[e], :] += H[src[e], :].  16 threads per edge, each owns a float4
// chunk (128-bit coalesced gather; 4 f32 atomics). Feature table (25.6 MB)
// is L2-resident, so both the gather and the atomics mostly hit L2.
// ---------------------------------------------------------------------------
__global__ __launch_bounds__(256)
void sage_scatter_kernel(const float* __restrict__ H,
                         const int* __restrict__ src,
                         const int* __restrict__ dst,
                         float* __restrict__ agg, int n_edges) {
    int t = blockIdx.x * blockDim.x + threadIdx.x;
    int e = t >> 4;                  // edge index
    int c = (t & 15) << 2;           // feature chunk [c, c+4)
    if (e >= n_edges) return;
    int s = src[e];
    int d = dst[e];
    const float4 v = *reinterpret_cast<const float4*>(H + (size_t)s * FEAT + c);
    float* o = agg + (size_t)d * FEAT + c;
    atomicAdd(o + 0, v.x);
    atomicAdd(o + 1, v.y);
    atomicAdd(o + 2, v.z);
    atomicAdd(o + 3, v.w);
}

// ---------------------------------------------------------------------------
// One SAGE layer:
//   OUT = act( X @ Wself + (AGG / max(deg,1)) @ Wneigh + b )
//
// Block = 128 threads = 4 waves; block owns a 16-row node tile, wave w owns
// output columns [16w, 16w+16). K=64 swept in steps of 4 with
// V_WMMA_F32_16X16X4_F32, self and neigh products chained into one f32
// accumulator (32 WMMAs/wave). No divergence: EXEC is all-1s for WMMA.
//
// Per-lane fragment mapping (ISA 7.12.2, 32-bit):
//   A 16x4 : row = lane&15, K = 2*(lane>>4) + {0,1}
//   B 4x16 : col = lane&15, K = 2*(lane>>4) + {0,1}
//   C 16x16: col = lane&15, row = 8*(lane>>4) + vgpr
// ---------------------------------------------------------------------------
__global__ __launch_bounds__(128)
void sage_layer_wmma_kernel(const float* __restrict__ X,
                            const float* __restrict__ AGG,
                            const float* __restrict__ DEG,
                            const float* __restrict__ Wself,   // [64(in) x 64(out)]
                            const float* __restrict__ Wneigh,  // [64(in) x 64(out)]
                            const float* __restrict__ bias,    // [64]
                            float* __restrict__ OUT,
                            int n_nodes, int do_relu) {
    const int wave  = threadIdx.x >> 5;           // 0..3 -> N tile
    const int lane  = threadIdx.x & 31;
    const int row0  = blockIdx.x * 16;
    const int m     = lane & 15;                  // A row within tile
    const int khalf = lane >> 4;                  // 0 or 1
    const int kb    = khalf * 2;                  // K sub-offset within step
    const int ncol  = wave * 16 + m;              // B/C column (0..63)

    int row = row0 + m;
    if (row >= n_nodes) row = n_nodes - 1;        // clamp, no divergence

    const float* xr = X   + (size_t)row * FEAT;
    const float* ar = AGG + (size_t)row * FEAT;
    const float  rinv = 1.0f / fmaxf(DEG[row], 1.0f);

    v8f acc = {0.f, 0.f, 0.f, 0.f, 0.f, 0.f, 0.f, 0.f};

    #pragma unroll
    for (int k = 0; k < FEAT; k += 4) {
        v2f a_s, a_n, b_s, b_n;
        a_s[0] = xr[k + kb];
        a_s[1] = xr[k + kb + 1];
        a_n[0] = ar[k + kb]     * rinv;
        a_n[1] = ar[k + kb + 1] * rinv;
        b_s[0] = Wself [(size_t)(k + kb)     * FEAT + ncol];
        b_s[1] = Wself [(size_t)(k + kb + 1) * FEAT + ncol];
        b_n[0] = Wneigh[(size_t)(k + kb)     * FEAT + ncol];
        b_n[1] = Wneigh[(size_t)(k + kb + 1) * FEAT + ncol];
        // (neg_a, A, neg_b, B, c_mod, C, reuse_a, reuse_b)
        acc = __builtin_amdgcn_wmma_f32_16x16x4_f32(
            false, a_s, false, b_s, (short)0, acc, false, false);
        acc = __builtin_amdgcn_wmma_f32_16x16x4_f32(
            false, a_n, false, b_n, (short)0, acc, false, false);
    }

    const float bn    = bias[ncol];
    const int   mbase = row0 + khalf * 8;
    #pragma unroll
    for (int v = 0; v < 8; ++v) {
        int mm = mbase + v;
        if (mm >= n_nodes) mm = n_nodes - 1;      // 100000 % 16 == 0: never taken
        float val = acc[v] + bn;
        if (do_relu) val = fmaxf(val, 0.0f);
        OUT[(size_t)mm * FEAT + ncol] = val;
    }
}

// ---------------------------------------------------------------------------
// Host-side launch (graph-capture safe: only kernel launches on `stream`).
// ---------------------------------------------------------------------------
extern "C" void kernel_launch(void* const* d_in, const int* in_sizes, int n_in,
                              void* d_out, int out_size, void* d_ws, size_t ws_size,
                              hipStream_t stream) {
    const float* x       = (const float*)d_in[0];
    const int*   src     = (const int*)  d_in[1];
    const int*   dst     = (const int*)  d_in[2];
    const float* W_self1 = (const float*)d_in[3];
    const float* W_neigh1= (const float*)d_in[4];
    const float* b1      = (const float*)d_in[5];
    const float* W_self2 = (const float*)d_in[6];
    const float* W_neigh2= (const float*)d_in[7];
    const float* b2      = (const float*)d_in[8];
    float*       out     = (float*)d_out;

    const long long feat_elems = (long long)N_NODES * FEAT;   // 6.4M floats
    float* agg = (float*)d_ws;                                // [N, 64]
    float* h1  = agg + feat_elems;                            // [N, 64]
    float* deg = h1  + feat_elems;                            // [N]

    const int zblk = 256, zgrid = 2048;
    const int scat_grid = (N_EDGES * 16 + 255) / 256;         // 16 thr/edge
    const int deg_grid  = (N_EDGES + 255) / 256;
    const int gemm_grid = (N_NODES + 15) / 16;                // 6250 tiles

    // ---- layer 1 ----
    sage_zero_kernel<<<zgrid, zblk, 0, stream>>>(agg, feat_elems);
    sage_zero_kernel<<<zgrid, zblk, 0, stream>>>(deg, (long long)N_NODES);
    sage_degree_kernel<<<deg_grid, 256, 0, stream>>>(dst, deg, N_EDGES);
    sage_scatter_kernel<<<scat_grid, 256, 0, stream>>>(x, src, dst, agg, N_EDGES);
    sage_layer_wmma_kernel<<<gemm_grid, 128, 0, stream>>>(
        x, agg, deg, W_self1, W_neigh1, b1, h1, N_NODES, /*relu=*/1);

    // ---- layer 2 (degree reused) ----
    sage_zero_kernel<<<zgrid, zblk, 0, stream>>>(agg, feat_elems);
    sage_scatter_kernel<<<scat_grid, 256, 0, stream>>>(h1, src, dst, agg, N_EDGES);
    sage_layer_wmma_kernel<<<gemm_grid, 128, 0, stream>>>(
        h1, agg, deg, W_self2, W_neigh2, b2, out, N_NODES, /*relu=*/0);
}